// PolicyGradientLossCombined_28260884807718
// MI455X (gfx1250) — compile-verified
//
#include <hip/hip_runtime.h>
#include <hip/hip_bf16.h>
#include <math.h>
#include <stdint.h>

// Problem constants (from reference)
#define M_DIM   512
#define E_DIM   8
#define K_TOP   10
#define NUM_MC  25
#define LAMBDA  1.0f
#define LOG_K_FACT 15.104412573075516f   // log(10!)

#define NWAVES  8
#define BLOCK   (NWAVES * 32)

// ---------------- address-space helper types (for gfx1250 async-to-LDS) ----
// Builtin signature (from hipcc diagnostic): (as1 int4* src, as3 int4* dst, Ii, Ii)
typedef int v4i __attribute__((ext_vector_type(4)));
typedef __attribute__((address_space(1))) v4i gv4i_t;   // global int4
typedef __attribute__((address_space(3))) v4i lv4i_t;   // LDS int4

#if defined(__gfx1250__) && \
    __has_builtin(__builtin_amdgcn_global_load_async_to_lds_b128) && \
    __has_builtin(__builtin_amdgcn_s_wait_asynccnt)
#define USE_ASYNC_LDS 1
#else
#define USE_ASYNC_LDS 0
#endif

// ---------------- small device helpers ------------------------------------
__device__ __forceinline__ uint32_t mix32(uint32_t x) {
    x ^= x >> 17; x *= 0xed5ad4bbu;
    x ^= x >> 11; x *= 0xac4c1b51u;
    x ^= x >> 15; x *= 0x31848babu;
    x ^= x >> 14;
    return x;
}

// Hardware transcendentals (v_log_f32 / v_exp_f32, TRANS pipe, co-issues
// with VALU per CDNA5 S_DELAY_ALU semantics). ~1 ulp: fine for an MC loss.
__device__ __forceinline__ float fast_log(float x) { return __logf(x); }
__device__ __forceinline__ float fast_exp(float x) { return __expf(x); }

__device__ __forceinline__ float wred_max(float v) {
#pragma unroll
    for (int o = 16; o; o >>= 1) v = fmaxf(v, __shfl_xor(v, o, 32));
    return v;
}
__device__ __forceinline__ float wred_sum(float v) {
#pragma unroll
    for (int o = 16; o; o >>= 1) v += __shfl_xor(v, o, 32);
    return v;
}

// ---------------- kernels ---------------------------------------------------
__global__ void pg_zero_kernel(float* out) { out[0] = 0.0f; }

__global__ __launch_bounds__(BLOCK)
void pg_loss_kernel(const float* __restrict__ score,
                    const float* __restrict__ relevance,
                    const float* __restrict__ eth,
                    float* __restrict__ out, int B)
{
    __shared__ float s_logp[M_DIM];
    __shared__ float s_rel [M_DIM];
    __shared__ float s_eth [M_DIM * E_DIM];   // 16 KB, async-staged
    __shared__ float s_red [NWAVES];
    __shared__ float s_part[NWAVES];
    __shared__ float s_bc;

    const int b    = blockIdx.x;
    const int tid  = threadIdx.x;
    const int lane = tid & 31;
    const int wid  = tid >> 5;

    // ---- kick off async staging of the 16KB eth row into LDS (overlaps
    //      with the softmax phase below) ----
    {
        const char* gsrc = (const char*)(eth + (size_t)b * (M_DIM * E_DIM));
#if USE_ASYNC_LDS
        const uint32_t lbase = (uint32_t)(uintptr_t)(void*)s_eth;
#pragma unroll
        for (int i = 0; i < (M_DIM * E_DIM * 4) / (BLOCK * 16); ++i) {
            const int c = tid + i * BLOCK;             // 16-byte chunk index
            gv4i_t* g = (gv4i_t*)(uintptr_t)(gsrc + c * 16);
            lv4i_t* l = (lv4i_t*)(lbase + c * 16);
            __builtin_amdgcn_global_load_async_to_lds_b128(g, l, 0, 0);
        }
#else
        float4*       ld = (float4*)s_eth;
        const float4* gs = (const float4*)gsrc;
        for (int c = tid; c < (M_DIM * E_DIM) / 4; c += BLOCK) ld[c] = gs[c];
#endif
    }

    // ---- phase A: softmax log-probs + relevance normalization ------------
    const float* srow = score     + (size_t)b * M_DIM;
    const float* rrow = relevance + (size_t)b * M_DIM;

    const float sc0 = srow[tid], sc1 = srow[tid + BLOCK];
    const float rl0 = rrow[tid], rl1 = rrow[tid + BLOCK];
    s_rel[tid] = rl0; s_rel[tid + BLOCK] = rl1;

    // block max
    float v = wred_max(fmaxf(sc0, sc1));
    if (lane == 0) s_red[wid] = v;
    __syncthreads();
    if (tid == 0) {
        float t = s_red[0];
        for (int i = 1; i < NWAVES; ++i) t = fmaxf(t, s_red[i]);
        s_bc = t;
    }
    __syncthreads();
    const float mx = s_bc;

    // block sum of exp
    const float e0 = fast_exp(sc0 - mx), e1 = fast_exp(sc1 - mx);
    v = wred_sum(e0 + e1);
    if (lane == 0) s_red[wid] = v;        // safe: barrier above separates reads
    __syncthreads();
    if (tid == 0) {
        float t = 0.f;
        for (int i = 0; i < NWAVES; ++i) t += s_red[i];
        s_bc = t;
    }
    __syncthreads();
    const float logZ = fast_log(s_bc);
    s_logp[tid]         = sc0 - mx - logZ;
    s_logp[tid + BLOCK] = sc1 - mx - logZ;

    // block sum of relevance
    v = wred_sum(rl0 + rl1);
    if (lane == 0) s_red[wid] = v;
    __syncthreads();
    if (tid == 0) {
        float t = 0.f;
        for (int i = 0; i < NWAVES; ++i) t += s_red[i];
        s_bc = t;
    }
    __syncthreads();
    const float relsum  = s_bc;
    const float inv_rel = (relsum > 0.f) ? 1.f / relsum : 0.f;
    const float totrel  = (relsum > 0.f) ? 1.f : 0.f;

#if USE_ASYNC_LDS
    __builtin_amdgcn_s_wait_asynccnt(0);   // each wave drains its own copies
#endif
    __syncthreads();                       // s_logp, s_rel, s_eth all visible

    // ---- phase B: per-(b,s) Gumbel top-K selection + loss terms ----------
    float wave_acc = 0.f;

    for (int s = wid; s < NUM_MC; s += NWAVES) {
        // 16 keys per lane, stride-32 (bank-conflict-free LDS reads)
        float key[M_DIM / 32];
        const uint32_t base = ((uint32_t)b * NUM_MC + (uint32_t)s) * M_DIM;
#pragma unroll
        for (int j = 0; j < M_DIM / 32; ++j) {
            const int mi = lane + j * 32;
            const uint32_t h = mix32((base + (uint32_t)mi) * 0x9E3779B1u + 0x85EBCA6Bu);
            const float u   = ((float)(h >> 8) + 0.5f) * (1.0f / 16777216.0f);
            const float gum = -fast_log(-fast_log(u));    // Gumbel(0,1)
            key[j] = s_logp[mi] + gum;
        }

        float logprob = 0.f;   // sum_k [ log p_k - log denom_k ]
        float denom   = 1.f;   // sum of all p == 1 by construction
        float Srel    = 0.f;   // raw relevance sum over top-K
        float ethacc  = 0.f;   // lane e<8: sum_k rel_k_raw * eth[m_k][e]

#pragma unroll 1
        for (int k = 0; k < K_TOP; ++k) {
            // local arg-max over 16 registers
            float bv = -INFINITY; int bm = 0x7fffffff;
#pragma unroll
            for (int j = 0; j < M_DIM / 32; ++j) {
                if (key[j] > bv) { bv = key[j]; bm = lane + j * 32; }
            }
            // wave32 arg-max reduction (ds_swizzle path)
#pragma unroll
            for (int o = 16; o; o >>= 1) {
                const float ov = __shfl_xor(bv, o, 32);
                const int   om = __shfl_xor(bm, o, 32);
                if (ov > bv || (ov == bv && om < bm)) { bv = ov; bm = om; }
            }
            // bm is the winning index on every lane; owner masks it out
            if ((bm & 31) == lane) key[bm >> 5] = -INFINITY;

            const float lp = s_logp[bm];         // LDS broadcast read
            const float pk = fast_exp(lp);
            logprob += lp - fast_log(fmaxf(denom, 1e-12f));
            denom   -= pk;

            const float rv = s_rel[bm];
            Srel += rv;
            if (lane < E_DIM) ethacc += rv * s_eth[bm * E_DIM + lane];
        }

        const float log_prob = logprob + LOG_K_FACT;
        const float delta    = 2.f * (Srel * inv_rel) - totrel;

        // entropy over E=8 features: pz = ethacc / sum(ethacc); the top-K
        // rel normalization cancels inside pz, so it is omitted.
        float dsum = ethacc;
#pragma unroll
        for (int o = 4; o; o >>= 1) dsum += __shfl_xor(dsum, o, 8);
        const float pz    = fmaxf(ethacc / fmaxf(dsum, 1e-30f), 1e-30f);
        float eterm = (lane < E_DIM) ? (-pz * fast_log(pz)) : 0.f;
#pragma unroll
        for (int o = 4; o; o >>= 1) eterm += __shfl_xor(eterm, o, 8);

        if (lane == 0) wave_acc += -log_prob * (delta + LAMBDA * eterm);
    }

    if (lane == 0) s_part[wid] = wave_acc;
    __syncthreads();
    if (tid == 0) {
        float t = 0.f;
#pragma unroll
        for (int i = 0; i < NWAVES; ++i) t += s_part[i];
        atomicAdd(out, t * (1.0f / ((float)NUM_MC * (float)B)));
    }
}

// ---------------- launcher --------------------------------------------------
extern "C" void kernel_launch(void* const* d_in, const int* in_sizes, int n_in,
                              void* d_out, int out_size, void* d_ws, size_t ws_size,
                              hipStream_t stream)
{
    const float* score     = (const float*)d_in[0];
    const float* relevance = (const float*)d_in[1];
    const float* eth       = (const float*)d_in[2];
    float* out             = (float*)d_out;

    const int B = in_sizes[0] / M_DIM;     // 2048

    pg_zero_kernel<<<dim3(1), dim3(1), 0, stream>>>(out);
    pg_loss_kernel<<<dim3(B), dim3(BLOCK), 0, stream>>>(score, relevance, eth, out, B);
    (void)n_in; (void)out_size; (void)d_ws; (void)ws_size;
}